// CubeHead_vanilla_68599217652234
// MI455X (gfx1250) — compile-verified
//
#include <hip/hip_runtime.h>
#include <hip/hip_bf16.h>
#include <stdint.h>

#define TPB 256
#define PSTRIDE 17  // 16 floats used per thread per buffer + 1 pad; odd => gcd(17,64)=1, bank-conflict-free

// Sutherland-Hodgman single-edge clip, exact reference semantics.
// in/out are per-thread LDS buffers (x,y interleaved).
__device__ __forceinline__ int clip_edge(const float* in, float* out, int cnt,
                                         float p1x, float p1y, float ex, float ey) {
  int ocnt = 0;
  for (int i = 0; i < cnt; ++i) {
    int j = (i + 1 == cnt) ? 0 : (i + 1);
    float cx = in[2 * i], cy = in[2 * i + 1];
    float nx = in[2 * j], ny = in[2 * j + 1];
    float dc = ex * (cy - p1y) - ey * (cx - p1x);
    float dn = ex * (ny - p1y) - ey * (nx - p1x);
    bool cin = dc >= 0.0f;
    bool nin = dn >= 0.0f;
    if (cin) { out[2 * ocnt] = cx; out[2 * ocnt + 1] = cy; ++ocnt; }
    if (cin != nin) {
      float den = dc - dn;
      float t = dc / (fabsf(den) > 1e-12f ? den : 1.0f);
      out[2 * ocnt]     = cx + t * (nx - cx);
      out[2 * ocnt + 1] = cy + t * (ny - cy);
      ++ocnt;
    }
  }
  return ocnt;
}

__global__ __launch_bounds__(TPB) void iou_kernel(
    const float* __restrict__ pc, const float* __restrict__ pd, const float* __restrict__ py,
    const float* __restrict__ gc, const float* __restrict__ gd, const float* __restrict__ gy,
    float* __restrict__ iou, int N) {
  __shared__ float sraw[8];                 // raw GT scalars staged via async-to-LDS
  __shared__ float sg[12];                  // [0..7]=4 BEV corners (x,z), 8=ylo, 9=yhi, 10=vol_g
  __shared__ float poly[2][TPB * PSTRIDE];  // per-thread ping-pong polygon buffers

  const int tid = threadIdx.x;
  const int g = blockIdx.y;

  // --- CDNA5 async global->LDS staging of this block's GT box (7 floats) ---
  if (tid < 7) {
    const float* src;
    if (tid < 3)      src = gc + 3 * g + tid;
    else if (tid < 6) src = gd + 3 * g + (tid - 3);
    else              src = gy + g;
    unsigned lds_off = (unsigned)(uintptr_t)(&sraw[tid]);
    uint64_t ga = (uint64_t)(uintptr_t)src;
    asm volatile("global_load_async_to_lds_b32 %0, %1, off"
                 :: "v"(lds_off), "v"(ga) : "memory");
  }
  asm volatile("s_wait_asynccnt 0" ::: "memory");
  __syncthreads();

  // --- thread 0 amortizes GT trig/corners over this block's 256 proposals ---
  if (tid == 0) {
    float cx = sraw[0], cyv = sraw[1], cz = sraw[2];
    float dxv = sraw[3], dyv = sraw[4], dzv = sraw[5];
    float yaw = sraw[6];
    float cs = cosf(yaw), sn = sinf(yaw);
    const float lxs[4] = {0.5f, -0.5f, -0.5f, 0.5f};
    const float lzs[4] = {0.5f, 0.5f, -0.5f, -0.5f};
#pragma unroll
    for (int k = 0; k < 4; ++k) {
      float ax = lxs[k] * dxv, az = lzs[k] * dzv;
      sg[2 * k]     = ax * cs - az * sn + cx;
      sg[2 * k + 1] = ax * sn + az * cs + cz;
    }
    sg[8]  = cyv - 0.5f * dyv;
    sg[9]  = cyv + 0.5f * dyv;
    sg[10] = dxv * dyv * dzv;
  }
  __syncthreads();

  const int p = blockIdx.x * TPB + tid;
  if (p >= N) return;

  float pcx = pc[3 * p], pcy = pc[3 * p + 1], pcz = pc[3 * p + 2];
  float pdx = pd[3 * p], pdy = pd[3 * p + 1], pdz = pd[3 * p + 2];
  float yaw = py[p];
  float cs = cosf(yaw), sn = sinf(yaw);

  float* b0 = &poly[0][tid * PSTRIDE];
  float* b1 = &poly[1][tid * PSTRIDE];

  // initial pred quad (CCW), local @ rot.T + center
  {
    const float lxs[4] = {0.5f, -0.5f, -0.5f, 0.5f};
    const float lzs[4] = {0.5f, 0.5f, -0.5f, -0.5f};
#pragma unroll
    for (int k = 0; k < 4; ++k) {
      float ax = lxs[k] * pdx, az = lzs[k] * pdz;
      b0[2 * k]     = ax * cs - az * sn + pcx;
      b0[2 * k + 1] = ax * sn + az * cs + pcz;
    }
  }

  int cnt = 4;
  float* bin = b0;
  float* bout = b1;
#pragma unroll
  for (int k = 0; k < 4; ++k) {
    float p1x = sg[2 * k], p1z = sg[2 * k + 1];
    int k2 = (k + 1) & 3;
    float ex = sg[2 * k2] - p1x, ez = sg[2 * k2 + 1] - p1z;
    cnt = clip_edge(bin, bout, cnt, p1x, p1z, ex, ez);
    float* t = bin; bin = bout; bout = t;
  }

  // shoelace area of clipped polygon (result lives in bin)
  float acc = 0.0f;
  for (int i = 0; i < cnt; ++i) {
    int j = (i + 1 == cnt) ? 0 : (i + 1);
    acc += bin[2 * i] * bin[2 * j + 1] - bin[2 * i + 1] * bin[2 * j];
  }
  float area = 0.5f * fabsf(acc);

  float ylo = fmaxf(pcy - 0.5f * pdy, sg[8]);
  float yhi = fminf(pcy + 0.5f * pdy, sg[9]);
  float yov = fmaxf(0.0f, yhi - ylo);
  float inter = area * yov;
  float volp = pdx * pdy * pdz;
  float denom = fmaxf(volp + sg[10] - inter, 1e-8f);
  iou[(size_t)g * N + p] = inter / denom;
}

// Per-GT-row reductions: first-occurrence argmax/max + prefix points {1,10,100,1000}.
__global__ __launch_bounds__(TPB) void reduce_kernel(
    const float* __restrict__ iou, int N, int M,
    float* __restrict__ bidx, float* __restrict__ biou,
    float* __restrict__ pmax, float* __restrict__ pidx) {
  const int g = blockIdx.x;
  const int tid = threadIdx.x;
  const float* row = iou + (size_t)g * N;

  float bv = -__builtin_huge_valf();
  int bi = 0x7fffffff;
  for (int j = tid; j < N; j += TPB) {
    float v = row[j];
    if (v > bv || (v == bv && j < bi)) { bv = v; bi = j; }
  }

  __shared__ float sv[TPB];
  __shared__ int si[TPB];
  sv[tid] = bv;
  si[tid] = bi;
  __syncthreads();
  for (int s = TPB / 2; s > 0; s >>= 1) {
    if (tid < s) {
      float v2 = sv[tid + s];
      int i2 = si[tid + s];
      if (v2 > sv[tid] || (v2 == sv[tid] && i2 < si[tid])) { sv[tid] = v2; si[tid] = i2; }
    }
    __syncthreads();
  }

  if (tid == 0) {
    biou[g] = sv[0];
    bidx[g] = (float)si[0];

    // prefix scan over first 1000 (L2-resident; strict > keeps first occurrence)
    float rv = row[0];
    int ri = 0;
    pmax[0 * M + g] = rv; pidx[0 * M + g] = 0.0f;
    for (int j = 1; j < 1000; ++j) {
      float v = row[j];
      if (v > rv) { rv = v; ri = j; }
      if (j == 9)  { pmax[1 * M + g] = rv; pidx[1 * M + g] = (float)ri; }
      if (j == 99) { pmax[2 * M + g] = rv; pidx[2 * M + g] = (float)ri; }
    }
    pmax[3 * M + g] = rv; pidx[3 * M + g] = (float)ri;
  }
}

extern "C" void kernel_launch(void* const* d_in, const int* in_sizes, int n_in,
                              void* d_out, int out_size, void* d_ws, size_t ws_size,
                              hipStream_t stream) {
  (void)n_in; (void)d_ws; (void)ws_size; (void)out_size;
  const float* pc = (const float*)d_in[0];
  const float* pd = (const float*)d_in[1];
  const float* py = (const float*)d_in[2];
  const float* gc = (const float*)d_in[3];
  const float* gd = (const float*)d_in[4];
  const float* gy = (const float*)d_in[5];
  const int N = in_sizes[0] / 3;  // 20000 proposals
  const int M = in_sizes[3] / 3;  // 64 GT boxes

  float* out  = (float*)d_out;
  float* iou  = out;                      // M*N
  float* bidx = out + (size_t)M * N;      // M
  float* biou = bidx + M;                 // M
  float* pmax = biou + M;                 // 4*M
  float* pidx = pmax + 4 * M;             // 4*M

  dim3 grid((N + TPB - 1) / TPB, M);
  iou_kernel<<<grid, TPB, 0, stream>>>(pc, pd, py, gc, gd, gy, iou, N);
  reduce_kernel<<<M, TPB, 0, stream>>>(iou, N, M, bidx, biou, pmax, pidx);
}